// simple_Net_10411000725656
// MI455X (gfx1250) — compile-verified
//
#include <hip/hip_runtime.h>
#include <stdint.h>

// Problem constants (from reference): T=1024, B=32768, I=1, H=3
#define T_STEPS 1024
#define B_BATCH 32768
#define CT      32     // timesteps per TDM tile
#define BLK     128    // threads per block == batch columns per block (4 wave32)
#define NBLK    (B_BATCH / BLK)   // 256 workgroups -> covers up to 256 WGPs

typedef uint32_t u32;
typedef u32 u32x4 __attribute__((ext_vector_type(4)));
typedef u32 u32x8 __attribute__((ext_vector_type(8)));

// Force a wave-uniform global value into an SGPR.
__device__ __forceinline__ float uload(const float* p) {
  return __uint_as_float((u32)__builtin_amdgcn_readfirstlane((int)__float_as_uint(*p)));
}

__device__ __forceinline__ float fast_tanh(float z) {
#if defined(__has_builtin) && __has_builtin(__builtin_amdgcn_tanhf)
  return __builtin_amdgcn_tanhf(z);           // gfx1250 v_tanh_f32
#else
  // tanh(z) = 1 - 2/(exp2(z*2*log2(e)) + 1)
  float e = __builtin_amdgcn_exp2f(2.885390081777927f * z);
  return 1.0f - 2.0f * __builtin_amdgcn_rcpf(e + 1.0f);
#endif
}

__device__ __forceinline__ float fast_sigmoid(float z) {
  return __builtin_fmaf(fast_tanh(0.5f * z), 0.5f, 0.5f);
}

// Issue a TDM 2D tile load: CT rows x BLK floats, row stride = B_BATCH floats.
// D# group0/group1 per CDNA5 ISA ch.8; groups 2/3 omitted (2D tensor).
__device__ __forceinline__ void tdm_load_tile(const float* gsrc, u32 lds_off) {
  uint64_t ga = (uint64_t)(uintptr_t)gsrc;
  u32x4 g0;
  g0[0] = 1u;                                            // count=1, user descriptor
  g0[1] = lds_off;                                       // lds_addr (bytes)
  g0[2] = (u32)ga;                                       // global_addr[31:0]
  g0[3] = (u32)((ga >> 32) & 0x1FFFFFFu) | (2u << 30);   // global_addr[56:32] | type=2
  u32x8 g1;
  g1[0] = 2u << 16;                                      // data_size = 4 bytes
  g1[1] = (u32)((BLK & 0xFFFF) << 16);                   // tensor_dim0[15:0]
  g1[2] = (u32)((BLK >> 16) | ((CT & 0xFFFF) << 16));    // tensor_dim0 hi | tensor_dim1 lo
  g1[3] = (u32)((CT >> 16) | ((BLK & 0xFFFF) << 16));    // tensor_dim1 hi | tile_dim0
  g1[4] = (u32)CT;                                       // tile_dim1 | tile_dim2=0
  g1[5] = (u32)B_BATCH;                                  // tensor_dim0_stride lo (elements)
  g1[6] = 0u;                                            // stride0 hi | stride1 lo
  g1[7] = 0u;                                            // stride1 hi
  asm volatile("tensor_load_to_lds %0, %1" :: "s"(g0), "s"(g1) : "memory");
}

__global__ __launch_bounds__(BLK) void lstm_fused_kernel(
    const float* __restrict__ x,    const float* __restrict__ h0,
    const float* __restrict__ c0,   const float* __restrict__ W_ih,
    const float* __restrict__ W_hh, const float* __restrict__ b_ih,
    const float* __restrict__ b_hh, float* __restrict__ ys) {
  __shared__ float xtile[2][CT * BLK];

  const int tid = threadIdx.x;
  const int b   = blockIdx.x * BLK + tid;

  // Uniform weights -> SGPRs. Gate order (PyTorch): rows 0-2=i, 3-5=f, 6-8=g, 9-11=o.
  float wih[12], bsum[12], whh[12][3];
#pragma unroll
  for (int g = 0; g < 12; ++g) {
    wih[g]  = uload(W_ih + g);
    bsum[g] = uload(b_ih + g) + uload(b_hh + g);
#pragma unroll
    for (int k = 0; k < 3; ++k) whh[g][k] = uload(W_hh + g * 3 + k);
  }

  float h[3], c[3];
#pragma unroll
  for (int j = 0; j < 3; ++j) { h[j] = h0[b * 3 + j]; c[j] = c0[b * 3 + j]; }

  const float* gbase = x + (size_t)blockIdx.x * BLK;
  const u32 lds0 = (u32)(uintptr_t)&xtile[0][0];
  const u32 lds1 = (u32)(uintptr_t)&xtile[1][0];
  const bool w0 = (tid < 32);   // wave 0 drives the DMA pipeline

  if (w0) {
    tdm_load_tile(gbase, lds0);
    tdm_load_tile(gbase + (size_t)CT * B_BATCH, lds1);
  }

  for (int ch = 0; ch < T_STEPS / CT; ++ch) {
    // <=1 outstanding => tile for chunk `ch` has landed (TDM is in-order per wave).
    if (w0) __builtin_amdgcn_s_wait_tensorcnt(1);
    __syncthreads();

    const float* xs = &xtile[ch & 1][0];
    float* out = ys + ((size_t)ch * CT * B_BATCH + (size_t)b) * 3;

#pragma unroll 4
    for (int tt = 0; tt < CT; ++tt) {
      float xv = xs[tt * BLK + tid];
      float pre[12];
#pragma unroll
      for (int g = 0; g < 12; ++g) {
        float a = __builtin_fmaf(xv, wih[g], bsum[g]);
        a = __builtin_fmaf(h[0], whh[g][0], a);
        a = __builtin_fmaf(h[1], whh[g][1], a);
        pre[g] = __builtin_fmaf(h[2], whh[g][2], a);
      }
#pragma unroll
      for (int j = 0; j < 3; ++j) {
        float ig = fast_sigmoid(pre[j]);
        float fg = fast_sigmoid(pre[3 + j]);
        float gg = fast_tanh(pre[6 + j]);
        float og = fast_sigmoid(pre[9 + j]);
        c[j] = __builtin_fmaf(fg, c[j], ig * gg);
        h[j] = og * fast_tanh(c[j]);
      }
      float* o = out + (size_t)tt * B_BATCH * 3;   // ys[t, b, :], streaming -> NT (merges to b96)
      __builtin_nontemporal_store(h[0], o + 0);
      __builtin_nontemporal_store(h[1], o + 1);
      __builtin_nontemporal_store(h[2], o + 2);
    }

    __syncthreads();   // everyone done reading xtile[ch&1]
    if (w0 && (ch + 2) < T_STEPS / CT) {
      tdm_load_tile(gbase + (size_t)(ch + 2) * CT * B_BATCH, (ch & 1) ? lds1 : lds0);
    }
  }
}

extern "C" void kernel_launch(void* const* d_in, const int* in_sizes, int n_in,
                              void* d_out, int out_size, void* d_ws, size_t ws_size,
                              hipStream_t stream) {
  (void)in_sizes; (void)n_in; (void)out_size; (void)d_ws; (void)ws_size;
  const float* x    = (const float*)d_in[0];
  const float* h0   = (const float*)d_in[1];
  const float* c0   = (const float*)d_in[2];
  const float* W_ih = (const float*)d_in[3];
  const float* W_hh = (const float*)d_in[4];
  const float* b_ih = (const float*)d_in[5];
  const float* b_hh = (const float*)d_in[6];
  float* ys = (float*)d_out;

  dim3 grid(NBLK);   // 256 blocks x 128 threads = 32768 lanes (one per sequence)
  lstm_fused_kernel<<<grid, BLK, 0, stream>>>(x, h0, c0, W_ih, W_hh, b_ih, b_hh, ys);
}